// Loss_197568496105
// MI455X (gfx1250) — compile-verified
//
#include <hip/hip_runtime.h>

typedef float v2f __attribute__((ext_vector_type(2)));
typedef float v8f __attribute__((ext_vector_type(8)));

#define BB    8
#define N1    2048
#define N2    2048
#define KPT   128
#define TH_C  0.1f
#define SL1B  0.1f
#define EPSV  1e-8f

// workspace layout (floats)
#define OFF_RM_PR 0                 // rowmin (dist)  pts<->recon_model : B*N1
#define OFF_CM_PR 16384             // colmin (dist^2) pts<->recon_model: B*N2
#define OFF_RM_PK 32768             // rowmin (dist)  pts<->kpt         : B*N1
#define OFF_CM_PK 49152             // colmin (dist^2) pts<->kpt        : B*KPT
#define OFF_ACC   50176             // [0]=diversity [1]=delta [2]=nocs [3]=pose
#define WS_FLOATS 50184

__device__ __forceinline__ float fsqrt(float x) { return __builtin_amdgcn_sqrtf(x); }

__device__ __forceinline__ float wave_sum(float v) {
    #pragma unroll
    for (int off = 16; off > 0; off >>= 1) v += __shfl_xor(v, off, 32);
    return v;
}

__global__ void init_ws_kernel(float* ws) {
    int i = blockIdx.x * blockDim.x + threadIdx.x;
    if (i < OFF_ACC)            ws[i] = __uint_as_float(0x7F800000u); // +inf
    else if (i < WS_FLOATS)     ws[i] = 0.0f;
}

// Bidirectional chamfer on squared distances (sqrt deferred — min is monotone).
// P = (B,N1,3), Q = (B,ncols,3).
// rowmin[b*N1+i]   = min_j dist(P_i,Q_j)      (actual distance; wave-private rows)
// colmin[b*ncols+j]= min_i dist^2(P_i,Q_j)    (squared; atomicMin on float bits >= 0)
// Each wave: one 16-row block, sweeps column tiles of 16, 4 WMMAs per iteration.
// Column atomic: both wave-halves min into the same colmin slot (no cross-lane DS
// traffic, no EXEC-divergent branch; non-returning atomics don't stall the wave).
__global__ void chamfer_wmma_kernel(const float* __restrict__ P,
                                    const float* __restrict__ Q,
                                    float* __restrict__ rowmin,
                                    unsigned int* __restrict__ colmin,
                                    int ncols) {
    const int lane = threadIdx.x & 31;
    const int wid  = (blockIdx.x * blockDim.x + threadIdx.x) >> 5;
    const int rbpb = N1 / 16;
    const int b    = wid / rbpb;
    const int r0   = (wid % rbpb) * 16;
    const int m    = lane & 15;
    const int hi   = lane >> 4;

    // A tile: 16 rows x K=4 (z-pad). ISA layout: lanes 0-15 -> K=0,1 ; lanes 16-31 -> K=2,3
    const float* pr = P + ((size_t)b * N1 + (r0 + m)) * 3;
    const float px = pr[0], py = pr[1], pz = pr[2];
    v2f a;
    a.x = hi ? pz   : px;
    a.y = hi ? 0.0f : py;
    const float mynorm = px * px + py * py + pz * pz;

    // lane needs ||p||^2 of rows r0 + r + 8*hi for r=0..7 (D-matrix M mapping)
    float pn2[8];
    #pragma unroll
    for (int r = 0; r < 8; ++r) pn2[r] = __shfl(mynorm, 8 * hi + r, 32);

    float rowrun2[8];
    #pragma unroll
    for (int r = 0; r < 8; ++r) rowrun2[r] = __builtin_inff();

    const float* qbase = Q + (size_t)b * ncols * 3;
    unsigned int* cbase = colmin + (size_t)b * ncols;
    const int ntiles = ncols >> 4;   // 128 or 8: divisible by 4
    for (int t = 0; t < ntiles; t += 4) {
        if (t + 4 < ntiles)
            __builtin_prefetch(qbase + (size_t)(((t + 4) << 4) + m) * 3, 0, 3);

        float qx[4], qy[4], qz[4];
        #pragma unroll
        for (int u = 0; u < 4; ++u) {
            const float* q = qbase + (size_t)(((t + u) << 4) + m) * 3;
            qx[u] = q[0]; qy[u] = q[1]; qz[u] = q[2];
        }

        #pragma unroll
        for (int u = 0; u < 4; ++u) {
            v2f bm;
            bm.x = hi ? qz[u] : qx[u];
            bm.y = hi ? 0.0f  : qy[u];
            const float qn2 = qx[u] * qx[u] + qy[u] * qy[u] + qz[u] * qz[u];

            v8f c = {};
            v8f d = __builtin_amdgcn_wmma_f32_16x16x4_f32(
                false, a, false, bm, (short)0, c, false, false);

            float cmin = __builtin_inff();
            #pragma unroll
            for (int r = 0; r < 8; ++r) {
                float e = fmaxf(pn2[r] + qn2 - 2.0f * d[r], 0.0f);
                rowrun2[r] = fminf(rowrun2[r], e);
                cmin = fminf(cmin, e);
            }
            // lanes l and l+16 both min into column N = m of this tile
            atomicMin(cbase + (((t + u) << 4) + m), __float_as_uint(cmin));
        }
    }

    // reduce rowrun2 over N (16 lanes within each half), sqrt once per row
    #pragma unroll
    for (int r = 0; r < 8; ++r) {
        float v = rowrun2[r];
        v = fminf(v, __shfl_xor(v, 8, 32));
        v = fminf(v, __shfl_xor(v, 4, 32));
        v = fminf(v, __shfl_xor(v, 2, 32));
        v = fminf(v, __shfl_xor(v, 1, 32));
        float dist = fsqrt(v);
        if (m == 0)
            rowmin[(size_t)b * N1 + r0 + r + 8 * hi] = dist;
    }
}

// diversity: mean over (B,K,K) of min(dist, TH), diagonal -> TH
__global__ void diversity_kernel(const float* __restrict__ kpt, float* acc) {
    int idx = blockIdx.x * blockDim.x + threadIdx.x; // < B*KPT*KPT
    int b   = idx / (KPT * KPT);
    int rem = idx % (KPT * KPT);
    int i = rem / KPT, j = rem % KPT;
    const float* pi = kpt + ((size_t)b * KPT + i) * 3;
    const float* pj = kpt + ((size_t)b * KPT + j) * 3;
    float dx = pi[0] - pj[0], dy = pi[1] - pj[1], dz = pi[2] - pj[2];
    float d2 = dx * dx + dy * dy + dz * dz;
    float v  = (i == j) ? TH_C : fminf(fsqrt(d2), TH_C);
    v = wave_sum(v);
    if ((threadIdx.x & 31) == 0) atomicAdd(acc, v);
}

// delta: sum of row norms of recon_delta (B*N1 rows)
__global__ void delta_kernel(const float* __restrict__ delta, float* acc) {
    int idx = blockIdx.x * blockDim.x + threadIdx.x; // < B*N1
    const float* p = delta + (size_t)idx * 3;
    float v = fsqrt(p[0] * p[0] + p[1] * p[1] + p[2] * p[2]);
    v = wave_sum(v);
    if ((threadIdx.x & 31) == 0) atomicAdd(acc, v);
}

// nocs smooth-L1: gt[k,j] = sum_i ((kpt[k,i]-t[i])/scale) * R[i,j]
__global__ void nocs_kernel(const float* __restrict__ kpt3d,
                            const float* __restrict__ prednocs,
                            const float* __restrict__ Rl,
                            const float* __restrict__ tl,
                            const float* __restrict__ sl,
                            float* acc) {
    int idx = blockIdx.x * blockDim.x + threadIdx.x; // < B*KPT
    int b = idx / KPT, k = idx % KPT;
    float s = fsqrt(sl[b * 3] * sl[b * 3] + sl[b * 3 + 1] * sl[b * 3 + 1] +
                    sl[b * 3 + 2] * sl[b * 3 + 2]) + EPSV;
    const float* p = kpt3d + ((size_t)b * KPT + k) * 3;
    float v0 = (p[0] - tl[b * 3 + 0]) / s;
    float v1 = (p[1] - tl[b * 3 + 1]) / s;
    float v2 = (p[2] - tl[b * 3 + 2]) / s;
    float sum = 0.0f;
    #pragma unroll
    for (int j = 0; j < 3; ++j) {
        float gt = v0 * Rl[b * 9 + 0 + j] + v1 * Rl[b * 9 + 3 + j] + v2 * Rl[b * 9 + 6 + j];
        float diff = fabsf(prednocs[((size_t)b * KPT + k) * 3 + j] - gt);
        sum += (diff > SL1B) ? (diff - 0.5f * SL1B) : (diff * diff / (2.0f * SL1B));
    }
    sum = wave_sum(sum);
    if ((threadIdx.x & 31) == 0) atomicAdd(acc, sum);
}

// pose: mean column-norms of dR + mean norms of dt, ds (pre-normalized here)
__global__ void pose_kernel(const float* Rp, const float* Rl,
                            const float* tp, const float* tl,
                            const float* sp, const float* sl, float* acc) {
    int b = threadIdx.x;
    float v = 0.0f;
    if (b < BB) {
        #pragma unroll
        for (int j = 0; j < 3; ++j) {
            float s2 = 0.0f;
            #pragma unroll
            for (int i = 0; i < 3; ++i) {
                float d = Rp[b * 9 + i * 3 + j] - Rl[b * 9 + i * 3 + j];
                s2 += d * d;
            }
            v += fsqrt(s2) * (1.0f / (3.0f * BB));
        }
        float dt0 = tp[b * 3] - tl[b * 3], dt1 = tp[b * 3 + 1] - tl[b * 3 + 1],
              dt2 = tp[b * 3 + 2] - tl[b * 3 + 2];
        v += fsqrt(dt0 * dt0 + dt1 * dt1 + dt2 * dt2) * (1.0f / BB);
        float ds0 = sp[b * 3] - sl[b * 3], ds1 = sp[b * 3 + 1] - sl[b * 3 + 1],
              ds2 = sp[b * 3 + 2] - sl[b * 3 + 2];
        v += fsqrt(ds0 * ds0 + ds1 * ds1 + ds2 * ds2) * (1.0f / BB);
    }
    v = wave_sum(v);
    if (threadIdx.x == 0) atomicAdd(acc, v);
}

__global__ void finalize_kernel(const float* __restrict__ ws, float* __restrict__ out) {
    const float w_pr     = 0.5f / (float)(BB * N1);   // N1 == N2
    const float w_pk_row = 0.5f / (float)(BB * N1);
    const float w_pk_col = 0.5f / (float)(BB * KPT);
    float s = 0.0f;
    // rowmin arrays already hold distances; colmin arrays hold squared distances
    for (int i = threadIdx.x; i < BB * N1;  i += blockDim.x) s += ws[OFF_RM_PR + i] * w_pr;
    for (int i = threadIdx.x; i < BB * N2;  i += blockDim.x) s += fsqrt(ws[OFF_CM_PR + i]) * w_pr;
    for (int i = threadIdx.x; i < BB * N1;  i += blockDim.x) s += ws[OFF_RM_PK + i] * w_pk_row;
    for (int i = threadIdx.x; i < BB * KPT; i += blockDim.x) s += fsqrt(ws[OFF_CM_PK + i]) * w_pk_col;
    s = wave_sum(s);
    __shared__ float sm[8];
    if ((threadIdx.x & 31) == 0) sm[threadIdx.x >> 5] = s;
    __syncthreads();
    if (threadIdx.x == 0) {
        float tot = 0.0f;
        for (int w = 0; w < (int)(blockDim.x >> 5); ++w) tot += sm[w];
        tot += ws[OFF_ACC + 0] / (float)(BB * KPT * KPT); // diversity
        tot += ws[OFF_ACC + 1] / (float)(BB * N1);        // delta
        tot += ws[OFF_ACC + 2] / (float)(BB * KPT);       // nocs
        tot += ws[OFF_ACC + 3];                           // pose
        out[0] = tot;
    }
}

extern "C" void kernel_launch(void* const* d_in, const int* in_sizes, int n_in,
                              void* d_out, int out_size, void* d_ws, size_t ws_size,
                              hipStream_t stream) {
    const float* pts    = (const float*)d_in[0];
    const float* rdelta = (const float*)d_in[1];
    const float* kpt3d  = (const float*)d_in[2];
    const float* rmodel = (const float*)d_in[3];
    const float* knocs  = (const float*)d_in[4];
    const float* Rp     = (const float*)d_in[5];
    const float* tp     = (const float*)d_in[6];
    const float* sp     = (const float*)d_in[7];
    const float* Rl     = (const float*)d_in[8];
    const float* tl     = (const float*)d_in[9];
    const float* sl     = (const float*)d_in[10];
    float* ws  = (float*)d_ws;
    float* out = (float*)d_out;

    if (ws_size < WS_FLOATS * sizeof(float)) return;

    init_ws_kernel<<<(WS_FLOATS + 255) / 256, 256, 0, stream>>>(ws);

    // pts <-> recon_model : 1024 waves (8 waves/block), 32 iters x 4 WMMA tiles
    chamfer_wmma_kernel<<<(BB * (N1 / 16)) / 8, 256, 0, stream>>>(
        pts, rmodel, ws + OFF_RM_PR, (unsigned int*)(ws + OFF_CM_PR), N2);
    // pts <-> pred_kpt_3d : same rows, 2 iters x 4 WMMA tiles
    chamfer_wmma_kernel<<<(BB * (N1 / 16)) / 8, 256, 0, stream>>>(
        pts, kpt3d, ws + OFF_RM_PK, (unsigned int*)(ws + OFF_CM_PK), KPT);

    diversity_kernel<<<(BB * KPT * KPT) / 256, 256, 0, stream>>>(kpt3d, ws + OFF_ACC + 0);
    delta_kernel<<<(BB * N1) / 256, 256, 0, stream>>>(rdelta, ws + OFF_ACC + 1);
    nocs_kernel<<<(BB * KPT) / 256, 256, 0, stream>>>(kpt3d, knocs, Rl, tl, sl, ws + OFF_ACC + 2);
    pose_kernel<<<1, 32, 0, stream>>>(Rp, Rl, tp, tl, sp, sl, ws + OFF_ACC + 3);

    finalize_kernel<<<1, 256, 0, stream>>>(ws, out);
}